// GAT_59261958750689
// MI455X (gfx1250) — compile-verified
//
#include <hip/hip_runtime.h>
#include <math.h>

#define N_NODES 50000
#define N_EDGES 400000
#define NW      512      // HEADS*HID = 4*128, output width of every GEMM
#define HID     128
#define BN_EPS  1e-5f

typedef float v2f __attribute__((ext_vector_type(2)));
typedef float v8f __attribute__((ext_vector_type(8)));

__device__ __forceinline__ float lrelu(float x) { return x > 0.f ? x : 0.2f * x; }

// ---------------- fill / copy ----------------
__global__ void fill_u32_k(unsigned* __restrict__ p, unsigned v, size_t n) {
    size_t i = (size_t)blockIdx.x * blockDim.x + threadIdx.x;
    if (i < n) p[i] = v;
}
__global__ void copy_u32_k(unsigned* __restrict__ dst, const unsigned* __restrict__ src, size_t n) {
    size_t i = (size_t)blockIdx.x * blockDim.x + threadIdx.x;
    if (i < n) dst[i] = src[i];
}

// ---------------- GEMM: H[M,512] = X[M,K] @ W[K,512], f32 WMMA 16x16x4 ------
// grid.x = M/16 (M=50000 -> 3125 exact), blockDim = 256 (8 waves).
// Wave w computes rows [blockIdx.x*16, +16) x cols [w*64, +64) as 4 WMMA tiles.
__global__ void gemm_wmma_f32(const float* __restrict__ X,
                              const float* __restrict__ W,
                              float* __restrict__ H, int K) {
    const int lane    = threadIdx.x & 31;
    const int wave    = threadIdx.x >> 5;
    const int rowBase = blockIdx.x << 4;
    const int colBase = wave << 6;
    const int halfSel = lane >> 4;       // 0: lanes 0-15, 1: lanes 16-31
    const int lmod    = lane & 15;
    const int arow    = rowBase + lmod;  // A: row per lane (dup across halves)
    const int kb0     = halfSel << 1;    // A/B: K offset 0 or 2
    const int c0      = colBase + lmod;  // B/C/D: col per lane

    v8f acc0 = {}, acc1 = {}, acc2 = {}, acc3 = {};
    const float* xrow = X + (size_t)arow * K;

    for (int k = 0; k < K; k += 4) {
        v2f a;
        a.x = xrow[k + kb0];
        a.y = xrow[k + kb0 + 1];
        const float* wp0 = W + (size_t)(k + kb0) * NW;
        const float* wp1 = wp0 + NW;
        v2f b0; b0.x = wp0[c0];      b0.y = wp1[c0];
        v2f b1; b1.x = wp0[c0 + 16]; b1.y = wp1[c0 + 16];
        v2f b2; b2.x = wp0[c0 + 32]; b2.y = wp1[c0 + 32];
        v2f b3; b3.x = wp0[c0 + 48]; b3.y = wp1[c0 + 48];
        acc0 = __builtin_amdgcn_wmma_f32_16x16x4_f32(false, a, false, b0, (short)0, acc0, false, false);
        acc1 = __builtin_amdgcn_wmma_f32_16x16x4_f32(false, a, false, b1, (short)0, acc1, false, false);
        acc2 = __builtin_amdgcn_wmma_f32_16x16x4_f32(false, a, false, b2, (short)0, acc2, false, false);
        acc3 = __builtin_amdgcn_wmma_f32_16x16x4_f32(false, a, false, b3, (short)0, acc3, false, false);
    }

    // C/D layout: VGPR v -> row rowBase + v + halfSel*8, col = lmod (+tile)
    const int srow = rowBase + (halfSel << 3);
    #pragma unroll
    for (int v = 0; v < 8; ++v) {
        float* hp = H + (size_t)(srow + v) * NW;
        hp[c0]      = acc0[v];
        hp[c0 + 16] = acc1[v];
        hp[c0 + 32] = acc2[v];
        hp[c0 + 48] = acc3[v];
    }
}

// ---------------- attention coefficients: als/ald [N,4] ----------------
// wave-per-node; channel c = j*32+lane (coalesced); head = j>>2.
__global__ void attn_coef_k(const float* __restrict__ H,
                            const float* __restrict__ asrc,
                            const float* __restrict__ adst,
                            float* __restrict__ als, float* __restrict__ ald,
                            int Nn) {
    const int lane = threadIdx.x & 31;
    const int n = blockIdx.x * (blockDim.x >> 5) + (threadIdx.x >> 5);
    if (n >= Nn) return;
    const float* hp = H + (size_t)n * NW;
    float ss[4] = {0.f, 0.f, 0.f, 0.f};
    float dd[4] = {0.f, 0.f, 0.f, 0.f};
    #pragma unroll
    for (int j = 0; j < 16; ++j) {
        int c = j * 32 + lane;
        float hv = hp[c];
        ss[j >> 2] += hv * asrc[c];
        dd[j >> 2] += hv * adst[c];
    }
    #pragma unroll
    for (int hd = 0; hd < 4; ++hd) {
        float vs = ss[hd], vd = dd[hd];
        for (int off = 16; off; off >>= 1) {
            vs += __shfl_xor(vs, off, 32);
            vd += __shfl_xor(vd, off, 32);
        }
        ss[hd] = vs; dd[hd] = vd;
    }
    if (lane == 0) {
        #pragma unroll
        for (int hd = 0; hd < 4; ++hd) {
            als[n * 4 + hd] = ss[hd];
            ald[n * 4 + hd] = dd[hd];
        }
    }
}

// ---------------- CSR build ----------------
// counts into rowptr[dst+1]
__global__ void csr_count_k(const int* __restrict__ ei, unsigned* __restrict__ rowptr, int E) {
    int e = blockIdx.x * blockDim.x + threadIdx.x;
    if (e >= E) return;
    atomicAdd(&rowptr[ei[E + e] + 1], 1u);
}

// in-place inclusive scan of n u32 values, single block of 1024 threads
__global__ void scan_u32_k(unsigned* __restrict__ data, int n) {
    __shared__ unsigned buf[1024];
    __shared__ unsigned carry;
    if (threadIdx.x == 0) carry = 0u;
    __syncthreads();
    for (int base = 0; base < n; base += 1024) {
        int i = base + (int)threadIdx.x;
        unsigned v = (i < n) ? data[i] : 0u;
        buf[threadIdx.x] = v;
        __syncthreads();
        for (int off = 1; off < 1024; off <<= 1) {
            unsigned t = (threadIdx.x >= (unsigned)off) ? buf[threadIdx.x - off] : 0u;
            __syncthreads();
            buf[threadIdx.x] += t;
            __syncthreads();
        }
        unsigned res = buf[threadIdx.x] + carry;
        if (i < n) data[i] = res;
        __syncthreads();                 // all reads of carry done
        if (threadIdx.x == 1023) carry = res;
        __syncthreads();                 // write visible before next chunk
    }
}

// bucket the source node of every edge by its destination
__global__ void csr_scatter_k(const int* __restrict__ ei, unsigned* __restrict__ cursor,
                              int* __restrict__ srcs, int E) {
    int e = blockIdx.x * blockDim.x + threadIdx.x;
    if (e >= E) return;
    int d = ei[E + e];
    unsigned pos = atomicAdd(&cursor[d], 1u);
    srcs[pos] = ei[e];
}

// ---------------- fused GAT aggregation (gather, no atomics) ----------------
// wave per destination node: segment softmax (max, sum) over incident edges,
// then register-accumulated weighted sum of h[src] over all 512 channels.
__global__ void gat_gather_k(const unsigned* __restrict__ rowptr,
                             const int* __restrict__ srcs,
                             const float* __restrict__ h,
                             const float* __restrict__ als,
                             const float* __restrict__ ald,
                             float* __restrict__ agg, int Nn) {
    const int lane = threadIdx.x & 31;
    const int n = blockIdx.x * (blockDim.x >> 5) + (threadIdx.x >> 5);
    if (n >= Nn) return;
    const unsigned beg = rowptr[n], end = rowptr[n + 1];
    const float4 ad = ((const float4*)ald)[n];

    // pass 1: per-head segment max of leaky_relu(als[src]+ald[n])
    float m[4] = {-INFINITY, -INFINITY, -INFINITY, -INFINITY};
    for (unsigned t = beg + lane; t < end; t += 32) {
        float4 as = ((const float4*)als)[srcs[t]];
        m[0] = fmaxf(m[0], lrelu(as.x + ad.x));
        m[1] = fmaxf(m[1], lrelu(as.y + ad.y));
        m[2] = fmaxf(m[2], lrelu(as.z + ad.z));
        m[3] = fmaxf(m[3], lrelu(as.w + ad.w));
    }
    #pragma unroll
    for (int hd = 0; hd < 4; ++hd)
        for (int off = 16; off; off >>= 1)
            m[hd] = fmaxf(m[hd], __shfl_xor(m[hd], off, 32));

    // pass 2: per-head segment sum of exp(e - m)
    float sum[4] = {0.f, 0.f, 0.f, 0.f};
    for (unsigned t = beg + lane; t < end; t += 32) {
        float4 as = ((const float4*)als)[srcs[t]];
        sum[0] += __expf(lrelu(as.x + ad.x) - m[0]);
        sum[1] += __expf(lrelu(as.y + ad.y) - m[1]);
        sum[2] += __expf(lrelu(as.z + ad.z) - m[2]);
        sum[3] += __expf(lrelu(as.w + ad.w) - m[3]);
    }
    #pragma unroll
    for (int hd = 0; hd < 4; ++hd) {
        for (int off = 16; off; off >>= 1)
            sum[hd] += __shfl_xor(sum[hd], off, 32);
        sum[hd] = 1.f / (sum[hd] + 1e-16f);   // inv normalizer
    }

    // pass 3: acc[c] = sum_e alpha[e,head(c)] * h[src_e, c]; c = j*32+lane
    float acc[16];
    #pragma unroll
    for (int j = 0; j < 16; ++j) acc[j] = 0.f;
    for (unsigned t = beg; t < end; ++t) {
        int s = srcs[t];                                  // uniform per wave
        float4 as = ((const float4*)als)[s];
        float alpha[4];
        alpha[0] = __expf(lrelu(as.x + ad.x) - m[0]) * sum[0];
        alpha[1] = __expf(lrelu(as.y + ad.y) - m[1]) * sum[1];
        alpha[2] = __expf(lrelu(as.z + ad.z) - m[2]) * sum[2];
        alpha[3] = __expf(lrelu(as.w + ad.w) - m[3]) * sum[3];
        const float* hp = h + (size_t)s * NW;
        #pragma unroll
        for (int j = 0; j < 16; ++j)
            acc[j] += alpha[j >> 2] * hp[j * 32 + lane];  // coalesced 128B
    }
    float* ap = agg + (size_t)n * NW;
    #pragma unroll
    for (int j = 0; j < 16; ++j) ap[j * 32 + lane] = acc[j];
}

// ---------------- mean over heads + bias ----------------
__global__ void head_mean_bias_k(const float* __restrict__ agg,
                                 const float* __restrict__ bias,
                                 float* __restrict__ out, int total) {
    int idx = blockIdx.x * blockDim.x + threadIdx.x;
    if (idx >= total) return;
    int n = idx >> 7, c = idx & 127;
    const float* ap = agg + (size_t)n * NW + c;
    out[idx] = 0.25f * (ap[0] + ap[128] + ap[256] + ap[384]) + bias[c];
}

// ---------------- batchnorm stats (sum, sumsq per channel) ----------------
__global__ void bn_stats_k(const float* __restrict__ x,
                           float* __restrict__ stats, int total) {
    __shared__ float s_sum[HID], s_sq[HID];
    int t = threadIdx.x;
    if (t < HID) { s_sum[t] = 0.f; s_sq[t] = 0.f; }
    __syncthreads();
    for (int idx = blockIdx.x * blockDim.x + t; idx < total;
         idx += gridDim.x * blockDim.x) {
        float v = x[idx];
        int c = idx & 127;
        unsafeAtomicAdd(&s_sum[c], v);
        unsafeAtomicAdd(&s_sq[c], v * v);
    }
    __syncthreads();
    if (t < HID) {
        unsafeAtomicAdd(&stats[t], s_sum[t]);
        unsafeAtomicAdd(&stats[HID + t], s_sq[t]);
    }
}

// ---------------- batchnorm apply + relu (in place) ----------------
__global__ void bn_apply_relu_k(float* __restrict__ x,
                                const float* __restrict__ stats,
                                const float* __restrict__ g,
                                const float* __restrict__ be,
                                int total, float invN) {
    int idx = blockIdx.x * blockDim.x + threadIdx.x;
    if (idx >= total) return;
    int c = idx & 127;
    float mu  = stats[c] * invN;
    float var = stats[HID + c] * invN - mu * mu;   // biased var, matches jnp.var
    float y = (x[idx] - mu) * rsqrtf(var + BN_EPS) * g[c] + be[c];
    x[idx] = y > 0.f ? y : 0.f;
}

// =======================================================================
extern "C" void kernel_launch(void* const* d_in, const int* in_sizes, int n_in,
                              void* d_out, int out_size, void* d_ws, size_t ws_size,
                              hipStream_t stream) {
    const float* xin[2]  = {(const float*)d_in[0], (const float*)d_in[2]};
    const int*   ei[2]   = {(const int*)d_in[1],   (const int*)d_in[3]};
    const float* W[3]    = {(const float*)d_in[4],  (const float*)d_in[8],  (const float*)d_in[12]};
    const float* asrc[3] = {(const float*)d_in[5],  (const float*)d_in[9],  (const float*)d_in[13]};
    const float* adst[3] = {(const float*)d_in[6],  (const float*)d_in[10], (const float*)d_in[14]};
    const float* bias[3] = {(const float*)d_in[7],  (const float*)d_in[11], (const float*)d_in[15]};
    const float* gam[2]  = {(const float*)d_in[16], (const float*)d_in[18]};
    const float* bet[2]  = {(const float*)d_in[17], (const float*)d_in[19]};

    // ---- carve workspace ----
    char* p = (char*)d_ws;
    auto take = [&](size_t bytes) {
        char* r = p;
        p += (bytes + 255) & ~(size_t)255;
        return (void*)r;
    };
    float*    h      = (float*)   take((size_t)N_NODES * NW * 4);     // 102.4 MB
    float*    agg    = (float*)   take((size_t)N_NODES * NW * 4);     // 102.4 MB
    float*    cur    = (float*)   take((size_t)N_NODES * HID * 4);    //  25.6 MB
    float*    als    = (float*)   take((size_t)N_NODES * 4 * 4);
    float*    ald    = (float*)   take((size_t)N_NODES * 4 * 4);
    unsigned* rowptr = (unsigned*)take((size_t)(N_NODES + 1) * 4);
    unsigned* cursor = (unsigned*)take((size_t)N_NODES * 4);
    int*      srcs   = (int*)     take((size_t)N_EDGES * 4);
    float*    stats  = (float*)   take(2 * HID * 4);

    const int TPB = 256;
    const int gGemm  = N_NODES / 16;                    // 3125 (exact)
    const int gNode8 = (N_NODES + 7) / 8;               // wave-per-node
    const int gEdgeT = (N_EDGES + TPB - 1) / TPB;       // thread-per-edge
    const int nc     = N_NODES * HID;                   // 6.4M
    const int gNC    = (nc + TPB - 1) / TPB;
    const float invN = 1.0f / (float)N_NODES;

    for (int g = 0; g < 2; ++g) {
        // ---- build CSR for this graph (amortized over 3 layers) ----
        fill_u32_k<<<(N_NODES + 1 + TPB - 1) / TPB, TPB, 0, stream>>>(rowptr, 0u, (size_t)(N_NODES + 1));
        csr_count_k<<<gEdgeT, TPB, 0, stream>>>(ei[g], rowptr, N_EDGES);
        scan_u32_k<<<1, 1024, 0, stream>>>(rowptr, N_NODES + 1);
        copy_u32_k<<<(N_NODES + TPB - 1) / TPB, TPB, 0, stream>>>(cursor, rowptr, (size_t)N_NODES);
        csr_scatter_k<<<gEdgeT, TPB, 0, stream>>>(ei[g], cursor, srcs, N_EDGES);

        for (int l = 0; l < 3; ++l) {
            const float* in = (l == 0) ? xin[g] : cur;
            const int K = (l == 0) ? 512 : HID;

            // h = in @ W[l]  (WMMA f32)
            gemm_wmma_f32<<<gGemm, TPB, 0, stream>>>(in, W[l], h, K);

            // attention logits per node/head
            attn_coef_k<<<gNode8, TPB, 0, stream>>>(h, asrc[l], adst[l], als, ald, N_NODES);

            // fused segment softmax + weighted gather (no atomics, plain stores)
            gat_gather_k<<<gNode8, TPB, 0, stream>>>(rowptr, srcs, h, als, ald, agg, N_NODES);

            // mean over heads + bias
            float* dst = (l == 2) ? ((float*)d_out + (size_t)g * nc) : cur;
            head_mean_bias_k<<<gNC, TPB, 0, stream>>>(agg, bias[l], dst, nc);

            // batchnorm + relu (layers 0,1 only)
            if (l < 2) {
                fill_u32_k<<<1, TPB, 0, stream>>>((unsigned*)stats, 0u, (size_t)(2 * HID));
                bn_stats_k<<<2048, TPB, 0, stream>>>(cur, stats, nc);
                bn_apply_relu_k<<<gNC, TPB, 0, stream>>>(cur, stats, gam[l], bet[l], nc, invN);
            }
        }
    }
}